// SingleRoIExtractor_55001351193186
// MI455X (gfx1250) — compile-verified
//
#include <hip/hip_runtime.h>
#include <hip/hip_bf16.h>
#include <stdint.h>

// ---------------------------------------------------------------------------
// FPN RoIAlign (SingleRoIExtractor, legacy aligned=False), MI455X / gfx1250.
// Memory-bound gather. CDNA5 path: per-lane async global->LDS tap-grid gather
// (GLOBAL_LOAD_ASYNC_TO_LDS_B32, ASYNCcnt) double-buffered across channels;
// all per-RoI weights pre-folded into registers so the hot loop is only
// async-issue -> s_wait_asynccnt -> ds_load+FMA -> coalesced store.
// ---------------------------------------------------------------------------

#define NSAMP   14      // OUT_SIZE * SAMPLE_NUM
#define NTAPS   28      // 2 taps per sample coordinate
#define GSZ     784     // 28 x 28 tap grid per channel (floats)
#define ROUNDS  25      // ceil(784 / 32) async b32 issues per wave per channel
#define TAILLN  16      // lanes active in last round: 784 - 24*32
#define CH      256
#define NBINS   49
#define NWAVES  8

// --- CDNA5-specific primitives (inline asm: portable across toolchains) -----
__device__ __forceinline__ void async_gather_b32(const float* g, unsigned lds_off) {
  // GLOBAL_LOAD_ASYNC_TO_LDS_B32: each lane copies 4 bytes from global address g to LDS offset lds_off; tracked by ASYNCcnt
  asm volatile("global_load_async_to_lds_b32 %0, %1, off" :: "v"(lds_off), "v"(g) : "memory");
}
__device__ __forceinline__ void wait_async_le25() {
  asm volatile("s_wait_asynccnt 25" ::: "memory");   // previous 25-issue batch retired
}
__device__ __forceinline__ void wait_async_le0() {
  asm volatile("s_wait_asynccnt 0" ::: "memory");
}
__device__ __forceinline__ void wait_ds0() {
  asm volatile("s_wait_dscnt 0" ::: "memory");       // WAR guard before overwriting a slab
}

__global__ __launch_bounds__(256) void roi_align_fpn_gfx1250(
    const float* __restrict__ f0, const float* __restrict__ f1,
    const float* __restrict__ f2, const float* __restrict__ f3,
    const float* __restrict__ rois, float* __restrict__ out)
{
  __shared__ int   s_xcol[NTAPS];          // x tap columns: [2i]=x0, [2i+1]=x1
  __shared__ int   s_yrow[NTAPS];          // y tap rows:    [2i]=y0, [2i+1]=y1
  __shared__ float s_lx[NSAMP], s_ly[NSAMP];
  __shared__ float s_vx[NSAMP], s_vy[NSAMP];
  __shared__ const float* s_cbase;         // &feat[lvl][b][0][0][0]
  __shared__ int   s_W, s_HW;
  __shared__ float s_G[NWAVES][2][GSZ];    // per-wave double-buffered tap grid

  const int tid = threadIdx.x;
  const int roi = blockIdx.x;
  const float* rp = rois + 5 * (size_t)roi;

  // ---- per-RoI setup (first wave) -----------------------------------------
  if (tid < 32) {
    const float bx1 = rp[1], by1 = rp[2], bx2 = rp[3], by2 = rp[4];
    const float scale = sqrtf((bx2 - bx1 + 1.0f) * (by2 - by1 + 1.0f));
    int lvl = (int)floorf(log2f(scale * (1.0f / 56.0f) + 1e-6f));
    lvl = lvl < 0 ? 0 : (lvl > 3 ? 3 : lvl);
    const int   Ht[4] = {200, 100, 50, 25};
    const int   Wt[4] = {336, 168, 84, 42};
    const float st[4] = {0.25f, 0.125f, 0.0625f, 0.03125f};
    const int H = Ht[lvl], W = Wt[lvl];
    const float sc = st[lvl];
    const float x1s = bx1 * sc, y1s = by1 * sc;
    const float rw = fmaxf(bx2 * sc - x1s, 1.0f);
    const float rh = fmaxf(by2 * sc - y1s, 1.0f);
    const float bw = rw * (1.0f / 7.0f), bh = rh * (1.0f / 7.0f);

    if (tid < NSAMP) {                      // x-axis sample taps
      const int i = tid;
      const float frac = (float)(i >> 1) + 0.25f + 0.5f * (float)(i & 1);
      const float x = x1s + frac * bw;
      s_vx[i] = (x > -1.0f && x < (float)W) ? 1.0f : 0.0f;
      const float xc = fminf(fmaxf(x, 0.0f), (float)(W - 1));
      const int ix0 = (int)floorf(xc);
      s_lx[i] = xc - (float)ix0;
      s_xcol[2 * i]     = ix0;
      s_xcol[2 * i + 1] = (ix0 + 1 < W) ? ix0 + 1 : W - 1;
    } else if (tid < 2 * NSAMP) {           // y-axis sample taps
      const int i = tid - NSAMP;
      const float frac = (float)(i >> 1) + 0.25f + 0.5f * (float)(i & 1);
      const float y = y1s + frac * bh;
      s_vy[i] = (y > -1.0f && y < (float)H) ? 1.0f : 0.0f;
      const float yc = fminf(fmaxf(y, 0.0f), (float)(H - 1));
      const int iy0 = (int)floorf(yc);
      s_ly[i] = yc - (float)iy0;
      s_yrow[2 * i]     = iy0;
      s_yrow[2 * i + 1] = (iy0 + 1 < H) ? iy0 + 1 : H - 1;
    } else if (tid == 31) {
      const int b = (int)rp[0];
      const float* base = (lvl == 0) ? f0 : (lvl == 1) ? f1 : (lvl == 2) ? f2 : f3;
      s_cbase = base + (size_t)b * CH * H * W;
      s_W = W;
      s_HW = H * W;
    }
  }
  __syncthreads();

  const int wave = tid >> 5;
  const int lane = tid & 31;
  const int W  = s_W;
  const int HW = s_HW;
  const float* cb0 = s_cbase;

  // Per-lane gather offsets within one channel's plane (same for all channels).
  int goff[ROUNDS];
#pragma unroll
  for (int r = 0; r < ROUNDS; ++r) {
    int e = lane + r * 32;
    int ee = (e < GSZ) ? e : 0;
    int row = ee / NTAPS;
    int col = ee - row * NTAPS;
    goff[r] = s_yrow[row] * W + s_xcol[col];
  }

  // Per-lane bin weights, fully pre-folded (validity * bilinear * 0.25 mean).
  // t=0: bin = lane (always valid); t=1: bin = lane + 32 (valid for lane < 17).
  const bool has1 = (lane + 32) < NBINS;
  int   rA[2][2], rB[2][2];     // [t][syl] G row offsets (floats)
  int   ccs[2][2];              // [t][sxl] G col offsets (floats)
  float wgt[2][2][2][4];        // [t][syl][sxl][tap]
#pragma unroll
  for (int t = 0; t < 2; ++t) {
    const int bin = (t == 0) ? lane : (has1 ? lane + 32 : 0);
    const int by  = bin / 7;
    const int bxi = bin - by * 7;
#pragma unroll
    for (int syl = 0; syl < 2; ++syl) {
      const int sy = 2 * by + syl;
      rA[t][syl] = (2 * sy) * NTAPS;
      rB[t][syl] = rA[t][syl] + NTAPS;
      const float lyv = s_ly[sy], hyv = 1.0f - lyv, vyv = s_vy[sy];
#pragma unroll
      for (int sxl = 0; sxl < 2; ++sxl) {
        const int sx = 2 * bxi + sxl;
        if (syl == 0) ccs[t][sxl] = 2 * sx;
        const float lxv = s_lx[sx], hxv = 1.0f - lxv;
        const float vv = 0.25f * vyv * s_vx[sx];   // fold 2x2-sample mean
        wgt[t][syl][sxl][0] = vv * hyv * hxv;
        wgt[t][syl][sxl][1] = vv * hyv * lxv;
        wgt[t][syl][sxl][2] = vv * lyv * hxv;
        wgt[t][syl][sxl][3] = vv * lyv * lxv;
      }
    }
  }

  float* g0 = &s_G[wave][0][0];
  float* g1 = &s_G[wave][1][0];
  const unsigned l0 = (unsigned)(size_t)g0;   // LDS byte offset of slab 0
  const unsigned l1 = (unsigned)(size_t)g1;

  const int c0 = wave * 32;                   // this wave's channel block
  float* op0 = out + ((size_t)roi * CH + c0) * NBINS + lane;        // t=0 stream
  float* op1 = op0 + 32;                                            // t=1 stream

  // Prologue: stage channel c0 into slab 0.
  {
    const float* cb = cb0 + (size_t)c0 * HW;
#pragma unroll
    for (int r = 0; r < ROUNDS; ++r)
      if (r < ROUNDS - 1 || lane < TAILLN)
        async_gather_b32(cb + goff[r], l0 + 4u * (unsigned)(lane + r * 32));
  }

  for (int k = 0; k < 32; ++k) {
    const unsigned lnext = (k & 1) ? l0 : l1;
    const float*   Gc    = (k & 1) ? g1 : g0;

    if (k + 1 < 32) {
      // Overwrites the slab whose ds reads finished last iteration; guard WAR.
      wait_ds0();
      const float* cb = cb0 + (size_t)(c0 + k + 1) * HW;
#pragma unroll
      for (int r = 0; r < ROUNDS; ++r)
        if (r < ROUNDS - 1 || lane < TAILLN)
          async_gather_b32(cb + goff[r], lnext + 4u * (unsigned)(lane + r * 32));
      wait_async_le25();                      // batch k fully landed in LDS
    } else {
      wait_async_le0();
    }

    // ---- interpolate this lane's bin(s) from the staged tap grid ----------
#pragma unroll
    for (int t = 0; t < 2; ++t) {
      if (t == 0 || has1) {
        float acc = 0.0f;
#pragma unroll
        for (int syl = 0; syl < 2; ++syl) {
#pragma unroll
          for (int sxl = 0; sxl < 2; ++sxl) {
            const int cc = ccs[t][sxl];
            const float* pA = Gc + rA[t][syl] + cc;
            const float* pB = Gc + rB[t][syl] + cc;
            acc += wgt[t][syl][sxl][0] * pA[0] + wgt[t][syl][sxl][1] * pA[1]
                 + wgt[t][syl][sxl][2] * pB[0] + wgt[t][syl][sxl][3] * pB[1];
          }
        }
        if (t == 0) op0[(size_t)k * NBINS] = acc;
        else        op1[(size_t)k * NBINS] = acc;
      }
    }
  }
}

extern "C" void kernel_launch(void* const* d_in, const int* in_sizes, int n_in,
                              void* d_out, int out_size, void* d_ws, size_t ws_size,
                              hipStream_t stream) {
  (void)n_in; (void)out_size; (void)d_ws; (void)ws_size;
  const float* f0   = (const float*)d_in[0];
  const float* f1   = (const float*)d_in[1];
  const float* f2   = (const float*)d_in[2];
  const float* f3   = (const float*)d_in[3];
  const float* rois = (const float*)d_in[4];
  const int K = in_sizes[4] / 5;              // number of RoIs
  if (K <= 0) return;
  roi_align_fpn_gfx1250<<<K, 256, 0, stream>>>(f0, f1, f2, f3, rois, (float*)d_out);
}